// Flashback_17738214933144
// MI455X (gfx1250) — compile-verified
//
#include <hip/hip_runtime.h>
#include <hip/hip_bf16.h>

// ---------------- problem constants (from reference) ----------------
#define S_    50
#define U_    128
#define N_    20002
#define H_    256
#define UC_   5000
#define NNZG  200000
#define NNZI  100000

// ---------------- WMMA types (CDNA5, wave32) ----------------
typedef __attribute__((ext_vector_type(2))) float v2f;   // A/B frag of 16x16x4 f32 WMMA
typedef __attribute__((ext_vector_type(8))) float v8f;   // C/D frag (16x16 f32)

// ---------------- gfx1250 async global->LDS support (probed) ----------------
#if defined(__gfx1250__) && __has_builtin(__builtin_amdgcn_global_load_async_to_lds_b128)
#define HAVE_ASYNC 1
#else
#define HAVE_ASYNC 0
#endif

#if HAVE_ASYNC
// builtin signature (from clang diagnostic): (v4i AS1* src, v4i AS3* dst, imm offset, imm cpol)
typedef __attribute__((__vector_size__(4 * sizeof(int)))) int v4i;
typedef __attribute__((address_space(1))) v4i gv4i;   // global (__device__)
typedef __attribute__((address_space(3))) v4i lv4i;   // LDS (__shared__)
// memory -> LDS, 16B per lane, tracked by ASYNCcnt (no VGPR round trip).
// Flat LDS addresses carry the LDS offset in the low 32 bits (aperture rule),
// so integer-mediated addrspace casts are value-correct.
__device__ __forceinline__ void async_g2l_b128(const float* g, float* l) {
  __builtin_amdgcn_global_load_async_to_lds_b128(
      (gv4i*)(uintptr_t)g, (lv4i*)(uintptr_t)l, 0, 0);
}
#endif

__device__ __forceinline__ void wait_async0() {
#if HAVE_ASYNC
#if __has_builtin(__builtin_amdgcn_s_wait_asynccnt)
  __builtin_amdgcn_s_wait_asynccnt(0);
#else
  asm volatile("s_wait_asynccnt 0x0" ::: "memory");
#endif
#endif
}

// ======================================================================
// Generic fp32 WMMA GEMM:  C[M,N] = A[M,K] @ B[K,N] (+bias) (+relu)
// Row-major. Call-site guarantees: M%64==0, K%16==0, lda%4==0.
// Block: 256 threads = 8 waves, tile BM=64 x BN=128, Kc=16.
// Double-buffered LDS; fast path stages tiles with async b128 loads
// (GLOBAL_LOAD_ASYNC_TO_LDS_B128), waits via s_wait_asynccnt.
// Wave (wm,wn) owns a 32x32 subtile = 2x2 of 16x16 WMMA accumulators.
// A frag: lane l, vgpr v -> A[M=l%16][K=2*(l/16)+v]
// B frag: lane l, vgpr v -> B[K=2*(l/16)+v][N=l%16]
// C/D:    lane l, vgpr g -> C[M=g+8*(l/16)][N=l%16]
// ======================================================================
#define BM 64
#define BN 128
#define KC 16
#define KPAD (KC + 4)   // A row stride 80B: 16B-aligned for b128, banks m*20+k all distinct

__global__ __launch_bounds__(256) void gemm_wmma_f32(
    const float* __restrict__ A, int lda,
    const float* __restrict__ B, int ldb,
    float*       __restrict__ C, int ldc,
    const float* __restrict__ bias,
    int M, int N, int K, int relu)
{
  __shared__ float As[2][BM][KPAD];
  __shared__ float Bs[2][KC][BN + 4];   // row stride 528B: 16B multiple

  const int tid  = threadIdx.x;
  const int lane = tid & 31;
  const int wave = tid >> 5;
  const int half = lane >> 4;    // K pair {0,1} vs {2,3}
  const int l16  = lane & 15;    // M (A) / N (B,C) index
  const int wm   = wave & 1;     // 2 row-groups of 32
  const int wn   = wave >> 1;    // 4 col-groups of 32
  const int row0 = blockIdx.y * BM;
  const int col0 = blockIdx.x * BN;

  const bool bfast = ((ldb & 3) == 0) && (col0 + BN <= N);

  // ---- tile stager: global -> LDS buffer `buf` for k-panel k0 ----
  auto stage_tile = [&](int k0, int buf) {
    {   // A: 64x16, one b128 per thread; always 16B aligned (lda%4==0, k0%16==0)
      int r = tid >> 2, c = (tid & 3) << 2;
      const float* gp = A + (size_t)(row0 + r) * lda + k0 + c;
      float* lp = &As[buf][r][c];
#if HAVE_ASYNC
      async_g2l_b128(gp, lp);
#else
      const float4 av = *(const float4*)gp;
      lp[0] = av.x; lp[1] = av.y; lp[2] = av.z; lp[3] = av.w;
#endif
    }
    if (bfast) {   // B: 16x128, two b128 per thread
#pragma unroll
      for (int c2 = 0; c2 < 2; ++c2) {
        int linear = tid + c2 * 256;
        int r = linear >> 5, cc = (linear & 31) << 2;
        const float* gp = B + (size_t)(k0 + r) * ldb + col0 + cc;
        float* lp = &Bs[buf][r][cc];
#if HAVE_ASYNC
        async_g2l_b128(gp, lp);
#else
        const float4 bv = *(const float4*)gp;
        lp[0] = bv.x; lp[1] = bv.y; lp[2] = bv.z; lp[3] = bv.w;
#endif
      }
    } else {       // misaligned ldb and/or N edge: guarded scalar staging
      if (k0 + KC < K)
        __builtin_prefetch(B + (size_t)(k0 + KC) * ldb + col0 + ((tid & 31) << 2), 0, 1);
#pragma unroll
      for (int c2 = 0; c2 < 2; ++c2) {
        int linear = tid + c2 * 256;
        int r = linear >> 5, cc = (linear & 31) << 2;
#pragma unroll
        for (int j = 0; j < 4; ++j) {
          int col = col0 + cc + j;
          Bs[buf][r][cc + j] = (col < N) ? B[(size_t)(k0 + r) * ldb + col] : 0.0f;
        }
      }
    }
  };

  v8f acc[2][2] = {};

  // prologue: stage first panel
  stage_tile(0, 0);
  wait_async0();
  __syncthreads();

  int cur = 0;
  for (int k0 = 0; k0 < K; k0 += KC) {
    const int nxt = cur ^ 1;
    if (k0 + KC < K) stage_tile(k0 + KC, nxt);   // overlap next panel with WMMA

    // ---- 4 WMMA k-steps of 4; uniform control flow, EXEC all-ones ----
#pragma unroll
    for (int kk = 0; kk < KC; kk += 4) {
      v2f a[2], b[2];
#pragma unroll
      for (int sm = 0; sm < 2; ++sm) {
        int m = wm * 32 + sm * 16 + l16;
        a[sm][0] = As[cur][m][kk + 2 * half + 0];
        a[sm][1] = As[cur][m][kk + 2 * half + 1];
      }
#pragma unroll
      for (int sn = 0; sn < 2; ++sn) {
        int n = wn * 32 + sn * 16 + l16;
        b[sn][0] = Bs[cur][kk + 2 * half + 0][n];
        b[sn][1] = Bs[cur][kk + 2 * half + 1][n];
      }
#pragma unroll
      for (int sm = 0; sm < 2; ++sm)
#pragma unroll
        for (int sn = 0; sn < 2; ++sn)
          acc[sm][sn] = __builtin_amdgcn_wmma_f32_16x16x4_f32(
              false, a[sm], false, b[sn], (short)0, acc[sm][sn], false, false);
    }

    wait_async0();      // next panel resident in LDS
    __syncthreads();    // all waves done reading `cur`, done writing `nxt`
    cur = nxt;
  }

  // ---- store C (+bias, +relu), guard N edge ----
#pragma unroll
  for (int sm = 0; sm < 2; ++sm) {
#pragma unroll
    for (int sn = 0; sn < 2; ++sn) {
      int col = col0 + wn * 32 + sn * 16 + l16;
      if (col < N) {
        float bv = bias ? bias[col] : 0.0f;
#pragma unroll
        for (int g = 0; g < 8; ++g) {
          int row = row0 + wm * 32 + sm * 16 + g + 8 * half;
          float v = acc[sm][sn][g] + bv;
          if (relu) v = fmaxf(v, 0.0f);
          C[(size_t)row * ldc + col] = v;
        }
      }
    }
  }
}

// ======================================================================
// Sparse segment-sum:  out[rows[e], :] += vals[e] * table[cols[e], :]
// ======================================================================
__global__ void scatter_add_k(const int* __restrict__ rows, const int* __restrict__ cols,
                              const float* __restrict__ vals, const float* __restrict__ table,
                              float* __restrict__ outt, int nnz)
{
  int idx = blockIdx.x * 256 + threadIdx.x;
  const int per = H_ / 4;  // 64
  if (idx >= nnz * per) return;
  int e  = idx / per;
  int h4 = (idx % per) * 4;
  float v = vals[e];
  int r = rows[e], c = cols[e];
  const float4 t = *(const float4*)(table + (size_t)c * H_ + h4);
  float* dst = outt + (size_t)r * H_ + h4;
  atomicAdd(dst + 0, v * t.x);
  atomicAdd(dst + 1, v * t.y);
  atomicAdd(dst + 2, v * t.z);
  atomicAdd(dst + 3, v * t.w);
}

// copy enc_table last 2 rows into encoder_weight (concat in reference)
__global__ void copy_tail_k(const float* __restrict__ enc_table, float* __restrict__ enc_w)
{
  int idx = blockIdx.x * 256 + threadIdx.x;
  if (idx < 2 * H_)
    enc_w[(size_t)(N_ - 2) * H_ + idx] = enc_table[(size_t)(N_ - 2) * H_ + idx];
}

// p_u[u,:] = user_table[active_user[u],:]
__global__ void pu_gather_k(const int* __restrict__ active, const float* __restrict__ user_table,
                            float* __restrict__ p_u)
{
  int u = blockIdx.x, h = threadIdx.x;
  p_u[(size_t)u * H_ + h] = user_table[(size_t)active[u] * H_ + h];
}

// h0[u,s,:] = enc_w[x[s,u],:]  and  sim[s,u] = exp(-||ew_u[active[u]] - x_emb||)
__global__ __launch_bounds__(256) void embed_sim_k(
    const int* __restrict__ x, const int* __restrict__ active,
    const float* __restrict__ enc_w, const float* __restrict__ ew_u,
    float* __restrict__ h0, float* __restrict__ sim)
{
  __shared__ float red[256];
  int u = blockIdx.x, sp = blockIdx.y, h = threadIdx.x;
  int loc = x[sp * U_ + u];
  float e = enc_w[(size_t)loc * H_ + h];
  h0[((size_t)u * S_ + sp) * H_ + h] = e;
  float d = ew_u[(size_t)active[u] * H_ + h] - e;
  red[h] = d * d;
  __syncthreads();
  for (int st = 128; st > 0; st >>= 1) { if (h < st) red[h] += red[h + st]; __syncthreads(); }
  if (h == 0) sim[sp * U_ + u] = expf(-sqrtf(red[0]));
}

// ======================================================================
// Causal attention per user: scores = q k^T / 16, softmax, attnv = attn @ v
// ======================================================================
__global__ __launch_bounds__(256) void attention_k(
    const float* __restrict__ q, const float* __restrict__ k,
    const float* __restrict__ v, float* __restrict__ attnv)
{
  extern __shared__ float smem[];
  float* vs = smem;              // S*H
  float* sc = smem + S_ * H_;    // S*S
  const int u = blockIdx.x, tid = threadIdx.x;
  const float* qu = q + (size_t)u * S_ * H_;
  const float* ku = k + (size_t)u * S_ * H_;
  const float* vu = v + (size_t)u * S_ * H_;

  for (int idx = tid; idx < S_ * H_; idx += 256) vs[idx] = vu[idx];
  for (int p = tid; p < S_ * S_; p += 256) {
    int i = p / S_, j = p % S_;
    float sco = -1e30f;
    if (j <= i) {
      float acc = 0.f;
      for (int h = 0; h < H_; ++h) acc += qu[i * H_ + h] * ku[j * H_ + h];
      sco = acc * (1.0f / 16.0f);   // 1/sqrt(H)
    }
    sc[p] = sco;
  }
  __syncthreads();
  if (tid < S_) {                    // softmax row tid (causal)
    int i = tid;
    float m = -1e30f;
    for (int j = 0; j <= i; ++j) m = fmaxf(m, sc[i * S_ + j]);
    float den = 0.f;
    for (int j = 0; j <= i; ++j) den += expf(sc[i * S_ + j] - m);
    float inv = 1.0f / den;
    for (int j = 0; j < S_; ++j)
      sc[i * S_ + j] = (j <= i) ? expf(sc[i * S_ + j] - m) * inv : 0.0f;
  }
  __syncthreads();
  int h = tid;                       // H == 256 == blockDim
  for (int i = 0; i < S_; ++i) {
    float acc = 0.f;
    for (int j = 0; j <= i; ++j) acc += sc[i * S_ + j] * vs[j * H_ + h];
    attnv[((size_t)u * S_ + i) * H_ + h] = acc;
  }
}

// o[row,:] = LayerNorm(a[row,:] + r[row,:]) * g + b    (one block per row)
__global__ __launch_bounds__(256) void resid_ln_k(
    const float* __restrict__ a, const float* __restrict__ r,
    const float* __restrict__ g, const float* __restrict__ b, float* __restrict__ o)
{
  __shared__ float red[256];
  size_t row = blockIdx.x;
  int h = threadIdx.x;
  float xv = a[row * H_ + h] + r[row * H_ + h];
  red[h] = xv; __syncthreads();
  for (int st = 128; st > 0; st >>= 1) { if (h < st) red[h] += red[h + st]; __syncthreads(); }
  float mu = red[0] * (1.0f / H_);
  __syncthreads();
  red[h] = xv * xv; __syncthreads();
  for (int st = 128; st > 0; st >>= 1) { if (h < st) red[h] += red[h + st]; __syncthreads(); }
  float var = red[0] * (1.0f / H_) - mu * mu;
  o[row * H_ + h] = (xv - mu) * rsqrtf(var + 1e-5f) * g[h] + b[h];
}

// out_w[i,u,:] = sum_j w[i,j,u] * h2[u,j,:] / sum_j w[i,j,u]
__global__ __launch_bounds__(256) void spatial_k(
    const float* __restrict__ s2, const float* __restrict__ sim,
    const float* __restrict__ h2, float* __restrict__ out_w)
{
  __shared__ float wj[S_];
  __shared__ float sumw;
  int u = blockIdx.x, i = blockIdx.y, tid = threadIdx.x;
  if (tid < S_) {
    int j = tid;
    float wv = 0.f;
    if (j <= i) {
      float dx = s2[(i * U_ + u) * 2 + 0] - s2[(j * U_ + u) * 2 + 0];
      float dy = s2[(i * U_ + u) * 2 + 1] - s2[(j * U_ + u) * 2 + 1];
      float dist = sqrtf(dx * dx + dy * dy);
      wv = (expf(-dist) + 1e-10f) * sim[j * U_ + u];   // LAMBDA_S = 1.0
    }
    wj[j] = wv;
  }
  __syncthreads();
  if (tid == 0) {
    float sm = 0.f;
    for (int j = 0; j < S_; ++j) sm += wj[j];
    sumw = sm;
  }
  __syncthreads();
  int h = tid;
  float acc = 0.f;
  for (int j = 0; j <= i; ++j) acc += wj[j] * h2[((size_t)u * S_ + j) * H_ + h];
  out_w[((size_t)(i * U_ + u)) * H_ + h] = acc / sumw;
}

// Ag[i,u,:] = concat(out_w[indexs[i,u], u, :], p_u[u,:])
__global__ void gather_concat_k(const int* __restrict__ indexs, const float* __restrict__ out_w,
                                const float* __restrict__ p_u, float* __restrict__ Ag, int total)
{
  int idx = blockIdx.x * 256 + threadIdx.x;
  if (idx >= total) return;
  int c  = idx & (2 * H_ - 1);   // 2H = 512
  int iu = idx >> 9;
  int u = iu % U_;
  int i = iu / U_;
  float val;
  if (c < H_) {
    int src = indexs[i * U_ + u];
    val = out_w[((size_t)(src * U_ + u)) * H_ + c];
  } else {
    val = p_u[(size_t)u * H_ + (c - H_)];
  }
  Ag[idx] = val;
}

// ======================================================================
extern "C" void kernel_launch(void* const* d_in, const int* in_sizes, int n_in,
                              void* d_out, int out_size, void* d_ws, size_t ws_size,
                              hipStream_t stream)
{
  const int*   x          = (const int*)  d_in[0];
  /* d_in[1] = x_adj, unused by reference */
  const int*   indexs     = (const int*)  d_in[2];
  const float* s2         = (const float*)d_in[3];
  const int*   active     = (const int*)  d_in[4];
  const int*   g_rows     = (const int*)  d_in[5];
  const int*   g_cols     = (const int*)  d_in[6];
  const float* g_vals     = (const float*)d_in[7];
  const int*   i_rows     = (const int*)  d_in[8];
  const int*   i_cols     = (const int*)  d_in[9];
  const float* i_vals     = (const float*)d_in[10];
  const float* enc_table  = (const float*)d_in[11];
  const float* user_table = (const float*)d_in[12];
  const float* Wq  = (const float*)d_in[13];
  const float* Wk  = (const float*)d_in[14];
  const float* Wv  = (const float*)d_in[15];
  const float* Wo  = (const float*)d_in[16];
  const float* ln1_g = (const float*)d_in[17];
  const float* ln1_b = (const float*)d_in[18];
  const float* ln2_g = (const float*)d_in[19];
  const float* ln2_b = (const float*)d_in[20];
  const float* w1  = (const float*)d_in[21];
  const float* b1  = (const float*)d_in[22];
  const float* w2  = (const float*)d_in[23];
  const float* b2  = (const float*)d_in[24];
  const float* fc_w = (const float*)d_in[25];
  const float* fc_b = (const float*)d_in[26];
  float* out = (float*)d_out;

  // -------- workspace layout (floats; all offsets 16B aligned) --------
  float* ws = (float*)d_ws;
  size_t off = 0;
  float* enc_w = ws + off; off += (size_t)N_ * H_;        // 5,120,512
  float* ew_u  = ws + off; off += (size_t)UC_ * H_;       // 1,280,000
  float* h0    = ws + off; off += (size_t)U_ * S_ * H_;
  float* q     = ws + off; off += (size_t)U_ * S_ * H_;
  float* k     = ws + off; off += (size_t)U_ * S_ * H_;
  float* v     = ws + off; off += (size_t)U_ * S_ * H_;
  float* attnv = ws + off; off += (size_t)U_ * S_ * H_;
  float* h1    = ws + off; off += (size_t)U_ * S_ * H_;
  float* ffn   = ws + off; off += (size_t)U_ * S_ * 4 * H_; // 6,553,600
  float* h2    = ws + off; off += (size_t)U_ * S_ * H_;
  float* out_w = ws + off; off += (size_t)S_ * U_ * H_;
  float* p_u   = ws + off; off += (size_t)U_ * H_;
  float* sim   = ws + off; off += (size_t)S_ * U_;
  float* Ag    = ws + off; off += (size_t)S_ * U_ * 2 * H_; // 3,276,800
  float* ao     = k;   // reuse: k dead after attention_k
  float* h2pre  = v;   // reuse: v dead after attention_k
  (void)ws_size; (void)in_sizes; (void)n_in; (void)out_size;

  const int M = U_ * S_;   // 6400, multiple of BM=64

  // zero the two atomic-accumulation targets every call (graph-safe memset nodes)
  (void)hipMemsetAsync(enc_w, 0, (size_t)N_ * H_ * sizeof(float), stream);
  (void)hipMemsetAsync(ew_u,  0, (size_t)UC_ * H_ * sizeof(float), stream);

  // sparse segment sums
  {
    int total = NNZG * (H_ / 4);
    scatter_add_k<<<(total + 255) / 256, 256, 0, stream>>>(g_rows, g_cols, g_vals, enc_table, enc_w, NNZG);
    total = NNZI * (H_ / 4);
    scatter_add_k<<<(total + 255) / 256, 256, 0, stream>>>(i_rows, i_cols, i_vals, enc_table, ew_u, NNZI);
  }
  copy_tail_k<<<2, 256, 0, stream>>>(enc_table, enc_w);
  pu_gather_k<<<U_, 256, 0, stream>>>(active, user_table, p_u);

  // x_emb gather (as h0 in (U,S,H)) + sim
  embed_sim_k<<<dim3(U_, S_), 256, 0, stream>>>(x, active, enc_w, ew_u, h0, sim);

  // QKV projections (WMMA)
  dim3 gproj((H_ + BN - 1) / BN, M / BM);
  gemm_wmma_f32<<<gproj, 256, 0, stream>>>(h0, H_, Wq, H_, q, H_, nullptr, M, H_, H_, 0);
  gemm_wmma_f32<<<gproj, 256, 0, stream>>>(h0, H_, Wk, H_, k, H_, nullptr, M, H_, H_, 0);
  gemm_wmma_f32<<<gproj, 256, 0, stream>>>(h0, H_, Wv, H_, v, H_, nullptr, M, H_, H_, 0);

  // causal attention per user
  size_t shmem = (size_t)(S_ * H_ + S_ * S_) * sizeof(float);  // 61,200 B
  attention_k<<<U_, 256, shmem, stream>>>(q, k, v, attnv);

  // Wo projection, residual + LN1
  gemm_wmma_f32<<<gproj, 256, 0, stream>>>(attnv, H_, Wo, H_, ao, H_, nullptr, M, H_, H_, 0);
  resid_ln_k<<<M, 256, 0, stream>>>(ao, h0, ln1_g, ln1_b, h1);

  // FFN: relu(h1 @ w1 + b1) @ w2 + b2, residual + LN2
  dim3 gffn1((4 * H_) / BN, M / BM);
  gemm_wmma_f32<<<gffn1, 256, 0, stream>>>(h1, H_, w1, 4 * H_, ffn, 4 * H_, b1, M, 4 * H_, H_, 1);
  gemm_wmma_f32<<<gproj, 256, 0, stream>>>(ffn, 4 * H_, w2, H_, h2pre, H_, b2, M, H_, 4 * H_, 0);
  resid_ln_k<<<M, 256, 0, stream>>>(h2pre, h1, ln2_g, ln2_b, h2);

  // spatial weighting
  spatial_k<<<dim3(U_, S_), 256, 0, stream>>>(s2, sim, h2, out_w);

  // gather rows per indexs + concat p_u  (take_along_axis pushed before final GEMM)
  int total = S_ * U_ * 2 * H_;
  gather_concat_k<<<(total + 255) / 256, 256, 0, stream>>>(indexs, out_w, p_u, Ag, total);

  // final 6400x20002x512 fp32 WMMA GEMM straight into d_out
  dim3 gfin((N_ + BN - 1) / BN, M / BM);
  gemm_wmma_f32<<<gfin, 256, 0, stream>>>(Ag, 2 * H_, fc_w, N_, out, N_, fc_b, M, N_, 2 * H_, 0);
}